// LocalMamba2D_51539607683
// MI455X (gfx1250) — compile-verified
//
#include <hip/hip_runtime.h>
#include <hip/hip_bf16.h>

// ---------------------------------------------------------------------------
// LocalMamba2D for MI455X (gfx1250, wave32).
// All GEMMs run on V_WMMA_F32_16X16X4_F32 (full fp32 precision; problem is
// memory-bound at ~22 GFLOP / ~0.5 GB traffic, so the f32 WMMA rate is ample).
// ---------------------------------------------------------------------------

typedef float v2f __attribute__((ext_vector_type(2)));
typedef float v8f __attribute__((ext_vector_type(8)));

#define CDIM    256
#define DIN     512
#define DTRANK  16
#define DSTATE  16
#define NWIN    288          // 8 * 6 * 6 windows
#define TOKENS  18432        // NWIN * 64
#define HW      2304         // 48*48
#define BATCH   8

static __device__ __forceinline__ v8f wmma4(v2f a, v2f b, v8f c) {
    // D = A(16x4,f32) * B(4x16,f32) + C(16x16,f32)
    return __builtin_amdgcn_wmma_f32_16x16x4_f32(
        /*neg_a=*/false, a, /*neg_b=*/false, b,
        /*c_mod=*/(short)0, c, /*reuse_a=*/false, /*reuse_b=*/false);
}

// ---------------------------------------------------------------------------
// K1: proj_in 1x1 conv as GEMM out[o,p] = sum_c W[o,c]*x[b,c,p], + bias,
//     scatter-stored directly into window/token-major layout t[token, o].
// ---------------------------------------------------------------------------
__global__ __launch_bounds__(256) void k1_proj_in(
    const float* __restrict__ x, const float* __restrict__ W,
    const float* __restrict__ bias, float* __restrict__ tbuf)
{
    const int wid  = threadIdx.x >> 5;
    const int lane = threadIdx.x & 31;
    const int tile = blockIdx.x * 8 + wid;              // 8 * 16 * 36 tiles
    const int bimg = tile / (16 * 36);
    const int rem  = tile - bimg * (16 * 36);
    const int mt   = rem / 36;
    const int nt   = rem - mt * 36;
    const int m_base = mt * 16;                         // o
    const int p_base = nt * 64;                         // pixel
    const int rrow = lane & 15;
    const int kh   = (lane >> 4) << 1;                  // 0 or 2
    const float* xb = x + (size_t)bimg * CDIM * HW;

    v8f acc[4] = {};
    for (int k0 = 0; k0 < CDIM; k0 += 4) {
        v2f a = *(const v2f*)(W + (size_t)(m_base + rrow) * CDIM + k0 + kh);
#pragma unroll
        for (int t = 0; t < 4; ++t) {
            const int p = p_base + t * 16 + rrow;
            v2f bf;
            bf.x = xb[(size_t)(k0 + kh)     * HW + p];
            bf.y = xb[(size_t)(k0 + kh + 1) * HW + p];
            acc[t] = wmma4(a, bf, acc[t]);
        }
    }
    const int moff = (lane >> 4) << 3;
#pragma unroll
    for (int t = 0; t < 4; ++t) {
        const int p = p_base + t * 16 + rrow;
        const int h = p / 48, w = p - h * 48;
        const int token = (bimg * 36 + (h >> 3) * 6 + (w >> 3)) * 64
                        + ((h & 7) << 3) + (w & 7);
#pragma unroll
        for (int r = 0; r < 8; ++r) {
            const int m = m_base + r + moff;
            tbuf[(size_t)token * CDIM + m] = acc[t][r] + bias[m];
        }
    }
}

// ---------------------------------------------------------------------------
// K2: LayerNorm over C=256 per token (in place).
// ---------------------------------------------------------------------------
__global__ __launch_bounds__(256) void k2_layernorm(
    float* __restrict__ t, const float* __restrict__ g, const float* __restrict__ b)
{
    const int row = blockIdx.x;
    const int c   = threadIdx.x;
    __shared__ float red[256];
    const float v = t[(size_t)row * CDIM + c];
    red[c] = v; __syncthreads();
    for (int s = 128; s > 0; s >>= 1) { if (c < s) red[c] += red[c + s]; __syncthreads(); }
    const float mean = red[0] * (1.0f / CDIM);
    __syncthreads();
    const float d = v - mean;
    red[c] = d * d; __syncthreads();
    for (int s = 128; s > 0; s >>= 1) { if (c < s) red[c] += red[c + s]; __syncthreads(); }
    const float var = red[0] * (1.0f / CDIM);
    t[(size_t)row * CDIM + c] = d * rsqrtf(var + 1e-5f) * g[c] + b[c];
}

// ---------------------------------------------------------------------------
// K3: in-projection xz[m,e] = sum_c t[m,c]*W_in[e,c]; split into xm / z.
// ---------------------------------------------------------------------------
__global__ __launch_bounds__(256) void k3_in_proj(
    const float* __restrict__ t, const float* __restrict__ Win,
    float* __restrict__ xm_raw, float* __restrict__ zbuf)
{
    const int wid  = threadIdx.x >> 5;
    const int lane = threadIdx.x & 31;
    const int tile = blockIdx.x * 8 + wid;              // (TOKENS/16) * 16 tiles
    const int mt = tile >> 4, nt = tile & 15;
    const int m_base = mt * 16, n_base = nt * 64;
    const int rrow = lane & 15, kh = (lane >> 4) << 1;

    v8f acc[4] = {};
    for (int k0 = 0; k0 < CDIM; k0 += 4) {
        v2f a = *(const v2f*)(t + (size_t)(m_base + rrow) * CDIM + k0 + kh);
#pragma unroll
        for (int q = 0; q < 4; ++q) {
            const int n = n_base + q * 16 + rrow;
            v2f bf = *(const v2f*)(Win + (size_t)n * CDIM + k0 + kh);
            acc[q] = wmma4(a, bf, acc[q]);
        }
    }
    const int moff = (lane >> 4) << 3;
#pragma unroll
    for (int q = 0; q < 4; ++q) {
        const int n = n_base + q * 16 + rrow;
#pragma unroll
        for (int r = 0; r < 8; ++r) {
            const int m = m_base + r + moff;
            if (n < DIN) xm_raw[(size_t)m * DIN + n]        = acc[q][r];
            else         zbuf  [(size_t)m * DIN + (n - DIN)] = acc[q][r];
        }
    }
}

// ---------------------------------------------------------------------------
// K4: depthwise causal conv1d (k=4) over L within window + SiLU.
// ---------------------------------------------------------------------------
__global__ __launch_bounds__(256) void k4_conv_silu(
    const float* __restrict__ xm_raw, const float* __restrict__ cw,
    const float* __restrict__ cb, float* __restrict__ xm_act)
{
    const size_t idx = (size_t)blockIdx.x * 256 + threadIdx.x;   // TOKENS*DIN
    const int d = (int)(idx & (DIN - 1));
    const size_t token = idx >> 9;
    const int l = (int)(token & 63);
    const size_t win = token >> 6;
    float s = cb[d];
#pragma unroll
    for (int j = 0; j < 4; ++j) {
        const int ls = l - 3 + j;
        if (ls >= 0)
            s = fmaf(cw[d * 4 + j], xm_raw[((win << 6) + ls) * DIN + d], s);
    }
    xm_act[idx] = s / (1.0f + __expf(-s));               // SiLU
}

// ---------------------------------------------------------------------------
// K5: x-projection xdb[m,e] = sum_d xm_act[m,d]*W_xproj[e,d], e in [0,48).
// ---------------------------------------------------------------------------
__global__ __launch_bounds__(256) void k5_xproj(
    const float* __restrict__ xm_act, const float* __restrict__ Wx,
    float* __restrict__ xdb)
{
    const int wid  = threadIdx.x >> 5;
    const int lane = threadIdx.x & 31;
    const int tile = blockIdx.x * 8 + wid;              // TOKENS/16 tiles
    const int m_base = tile * 16;
    const int rrow = lane & 15, kh = (lane >> 4) << 1;

    v8f acc[3] = {};
    for (int k0 = 0; k0 < DIN; k0 += 4) {
        v2f a = *(const v2f*)(xm_act + (size_t)(m_base + rrow) * DIN + k0 + kh);
#pragma unroll
        for (int q = 0; q < 3; ++q) {
            const int n = q * 16 + rrow;
            v2f bf = *(const v2f*)(Wx + (size_t)n * DIN + k0 + kh);
            acc[q] = wmma4(a, bf, acc[q]);
        }
    }
    const int moff = (lane >> 4) << 3;
#pragma unroll
    for (int q = 0; q < 3; ++q) {
        const int n = q * 16 + rrow;
#pragma unroll
        for (int r = 0; r < 8; ++r)
            xdb[(size_t)(m_base + r + moff) * 48 + n] = acc[q][r];
    }
}

// ---------------------------------------------------------------------------
// K6: dt[m,d] = softplus(sum_r xdb[m,r]*W_dt[d,r] + b_dt[d]),  K=16.
// ---------------------------------------------------------------------------
__global__ __launch_bounds__(256) void k6_dtproj(
    const float* __restrict__ xdb, const float* __restrict__ Wdt,
    const float* __restrict__ bdt, float* __restrict__ dtb)
{
    const int wid  = threadIdx.x >> 5;
    const int lane = threadIdx.x & 31;
    const int tile = blockIdx.x * 8 + wid;              // (TOKENS/16) * 8 tiles
    const int mt = tile >> 3, nt = tile & 7;
    const int m_base = mt * 16, n_base = nt * 64;
    const int rrow = lane & 15, kh = (lane >> 4) << 1;

    v8f acc[4] = {};
#pragma unroll
    for (int k0 = 0; k0 < DTRANK; k0 += 4) {
        v2f a = *(const v2f*)(xdb + (size_t)(m_base + rrow) * 48 + k0 + kh);
#pragma unroll
        for (int q = 0; q < 4; ++q) {
            const int n = n_base + q * 16 + rrow;
            v2f bf = *(const v2f*)(Wdt + (size_t)n * DTRANK + k0 + kh);
            acc[q] = wmma4(a, bf, acc[q]);
        }
    }
    const int moff = (lane >> 4) << 3;
#pragma unroll
    for (int q = 0; q < 4; ++q) {
        const int n = n_base + q * 16 + rrow;
        const float bv = bdt[n];
#pragma unroll
        for (int r = 0; r < 8; ++r) {
            const float v = acc[q][r] + bv;
            const float sp = (v > 20.0f) ? v : log1pf(__expf(v));
            dtb[(size_t)(m_base + r + moff) * DIN + n] = sp;
        }
    }
}

// ---------------------------------------------------------------------------
// K7: selective scan. One block per window; 256 threads x 2 channels each;
//     full 512x16 state in registers; B/C staged in LDS once (one barrier).
// ---------------------------------------------------------------------------
__global__ __launch_bounds__(256) void k7_scan(
    const float* __restrict__ xdb, const float* __restrict__ dtb,
    const float* __restrict__ xm_act, const float* __restrict__ Alog,
    float* __restrict__ ybuf)
{
    const int win = blockIdx.x;
    const int tid = threadIdx.x;
    __shared__ float sB[64 * DSTATE];
    __shared__ float sC[64 * DSTATE];
    for (int i = tid; i < 64 * DSTATE; i += 256) {
        const int l = i >> 4, n = i & 15;
        const size_t row = (size_t)(win * 64 + l) * 48;
        sB[i] = xdb[row + DTRANK + n];
        sC[i] = xdb[row + DTRANK + DSTATE + n];
    }
    const int d0 = tid * 2;
    float Ar[2][DSTATE], h[2][DSTATE];
#pragma unroll
    for (int j = 0; j < 2; ++j)
#pragma unroll
        for (int n = 0; n < DSTATE; ++n) {
            Ar[j][n] = -__expf(Alog[(size_t)(d0 + j) * DSTATE + n]);
            h[j][n]  = 0.0f;
        }
    __syncthreads();
    for (int l = 0; l < 64; ++l) {
        const size_t token = (size_t)win * 64 + l;
        const float dt0 = dtb[token * DIN + d0];
        const float dt1 = dtb[token * DIN + d0 + 1];
        const float x0  = xm_act[token * DIN + d0];
        const float x1  = xm_act[token * DIN + d0 + 1];
        const float u0 = dt0 * x0, u1 = dt1 * x1;
        const float* Bl = &sB[l * DSTATE];
        const float* Cl = &sC[l * DSTATE];
        float y0 = 0.0f, y1 = 0.0f;
#pragma unroll
        for (int n = 0; n < DSTATE; ++n) {
            const float bn = Bl[n], cn = Cl[n];
            const float dA0 = __expf(dt0 * Ar[0][n]);
            h[0][n] = fmaf(dA0, h[0][n], u0 * bn);
            y0 = fmaf(h[0][n], cn, y0);
            const float dA1 = __expf(dt1 * Ar[1][n]);
            h[1][n] = fmaf(dA1, h[1][n], u1 * bn);
            y1 = fmaf(h[1][n], cn, y1);
        }
        ybuf[token * DIN + d0]     = y0;
        ybuf[token * DIN + d0 + 1] = y1;
    }
}

// ---------------------------------------------------------------------------
// K8: gating: y = (y + xm*D) * silu(z)   (in place on y).
// ---------------------------------------------------------------------------
__global__ __launch_bounds__(256) void k8_gate(
    float* __restrict__ ybuf, const float* __restrict__ xm_act,
    const float* __restrict__ Dp, const float* __restrict__ zbuf)
{
    const size_t idx = (size_t)blockIdx.x * 256 + threadIdx.x;
    const int d = (int)(idx & (DIN - 1));
    const float v = fmaf(xm_act[idx], Dp[d], ybuf[idx]);
    const float z = zbuf[idx];
    ybuf[idx] = v * (z / (1.0f + __expf(-z)));
}

// ---------------------------------------------------------------------------
// K9a: out-projection ytok[m,c] = sum_d y[m,d]*W_out[c,d].
// ---------------------------------------------------------------------------
__global__ __launch_bounds__(256) void k9a_out_proj(
    const float* __restrict__ ybuf, const float* __restrict__ Wout,
    float* __restrict__ ytok)
{
    const int wid  = threadIdx.x >> 5;
    const int lane = threadIdx.x & 31;
    const int tile = blockIdx.x * 8 + wid;              // (TOKENS/16) * 4 tiles
    const int mt = tile >> 2, nt = tile & 3;
    const int m_base = mt * 16, n_base = nt * 64;
    const int rrow = lane & 15, kh = (lane >> 4) << 1;

    v8f acc[4] = {};
    for (int k0 = 0; k0 < DIN; k0 += 4) {
        v2f a = *(const v2f*)(ybuf + (size_t)(m_base + rrow) * DIN + k0 + kh);
#pragma unroll
        for (int q = 0; q < 4; ++q) {
            const int n = n_base + q * 16 + rrow;
            v2f bf = *(const v2f*)(Wout + (size_t)n * DIN + k0 + kh);
            acc[q] = wmma4(a, bf, acc[q]);
        }
    }
    const int moff = (lane >> 4) << 3;
#pragma unroll
    for (int q = 0; q < 4; ++q) {
        const int n = n_base + q * 16 + rrow;
#pragma unroll
        for (int r = 0; r < 8; ++r)
            ytok[(size_t)(m_base + r + moff) * CDIM + n] = acc[q][r];
    }
}

// ---------------------------------------------------------------------------
// K9b: window-reverse + proj_out 1x1 conv + bias + residual.
//      out[b,o,p] = sum_c Wpo[o,c]*ytok[token(b,p),c] + bpo[o] + x[b,o,p]
// ---------------------------------------------------------------------------
__global__ __launch_bounds__(256) void k9b_proj_out(
    const float* __restrict__ ytok, const float* __restrict__ Wpo,
    const float* __restrict__ bpo, const float* __restrict__ x,
    float* __restrict__ out)
{
    const int wid  = threadIdx.x >> 5;
    const int lane = threadIdx.x & 31;
    const int tile = blockIdx.x * 8 + wid;              // 8 * 16 * 36 tiles
    const int bimg = tile / (16 * 36);
    const int rem  = tile - bimg * (16 * 36);
    const int mt   = rem / 36;
    const int nt   = rem - mt * 36;
    const int m_base = mt * 16;                         // o
    const int p_base = nt * 64;                         // pixel
    const int rrow = lane & 15, kh = (lane >> 4) << 1;

    // per-lane token index for each of the 4 N-subtiles
    int tok[4];
#pragma unroll
    for (int q = 0; q < 4; ++q) {
        const int p = p_base + q * 16 + rrow;
        const int h = p / 48, w = p - h * 48;
        tok[q] = (bimg * 36 + (h >> 3) * 6 + (w >> 3)) * 64
               + ((h & 7) << 3) + (w & 7);
    }

    v8f acc[4] = {};
    for (int k0 = 0; k0 < CDIM; k0 += 4) {
        v2f a = *(const v2f*)(Wpo + (size_t)(m_base + rrow) * CDIM + k0 + kh);
#pragma unroll
        for (int q = 0; q < 4; ++q) {
            v2f bf = *(const v2f*)(ytok + (size_t)tok[q] * CDIM + k0 + kh);
            acc[q] = wmma4(a, bf, acc[q]);
        }
    }
    const int moff = (lane >> 4) << 3;
#pragma unroll
    for (int q = 0; q < 4; ++q) {
        const int p = p_base + q * 16 + rrow;
#pragma unroll
        for (int r = 0; r < 8; ++r) {
            const int m = m_base + r + moff;
            const size_t oi = ((size_t)bimg * CDIM + m) * HW + p;
            out[oi] = acc[q][r] + bpo[m] + x[oi];
        }
    }
}

// ---------------------------------------------------------------------------
extern "C" void kernel_launch(void* const* d_in, const int* in_sizes, int n_in,
                              void* d_out, int out_size, void* d_ws, size_t ws_size,
                              hipStream_t stream)
{
    const float* x       = (const float*)d_in[0];
    const float* piw     = (const float*)d_in[1];
    const float* pib     = (const float*)d_in[2];
    const float* ln_g    = (const float*)d_in[3];
    const float* ln_b    = (const float*)d_in[4];
    const float* W_in    = (const float*)d_in[5];
    const float* conv_w  = (const float*)d_in[6];
    const float* conv_b  = (const float*)d_in[7];
    const float* W_xproj = (const float*)d_in[8];
    const float* W_dt    = (const float*)d_in[9];
    const float* b_dt    = (const float*)d_in[10];
    const float* A_log   = (const float*)d_in[11];
    const float* Dp      = (const float*)d_in[12];
    const float* W_out   = (const float*)d_in[13];
    const float* pow_    = (const float*)d_in[14];
    const float* pob     = (const float*)d_in[15];
    float* out = (float*)d_out;

    // Workspace layout (floats). Total = TOKENS*(256+512*4+48) ~= 173 MB.
    float* ws      = (float*)d_ws;
    const size_t TOK = TOKENS;
    float* t_buf   = ws;                        // TOK*256  (reused as ytok)
    float* xm_raw  = t_buf  + TOK * CDIM;       // TOK*512  (reused as dt)
    float* zbuf    = xm_raw + TOK * DIN;        // TOK*512
    float* xm_act  = zbuf   + TOK * DIN;        // TOK*512
    float* xdbb    = xm_act + TOK * DIN;        // TOK*48
    float* ybuf    = xdbb   + TOK * 48;         // TOK*512
    float* dtb     = xm_raw;                    // alias: xm_raw dead after K4
    float* ytok    = t_buf;                     // alias: t dead after K3

    k1_proj_in   <<<576,   256, 0, stream>>>(x, piw, pib, t_buf);
    k2_layernorm <<<TOKENS,256, 0, stream>>>(t_buf, ln_g, ln_b);
    k3_in_proj   <<<2304,  256, 0, stream>>>(t_buf, W_in, xm_raw, zbuf);
    k4_conv_silu <<<36864, 256, 0, stream>>>(xm_raw, conv_w, conv_b, xm_act);
    k5_xproj     <<<144,   256, 0, stream>>>(xm_act, W_xproj, xdbb);
    k6_dtproj    <<<1152,  256, 0, stream>>>(xdbb, W_dt, b_dt, dtb);
    k7_scan      <<<NWIN,  256, 0, stream>>>(xdbb, dtb, xm_act, A_log, ybuf);
    k8_gate      <<<36864, 256, 0, stream>>>(ybuf, xm_act, Dp, zbuf);
    k9a_out_proj <<<576,   256, 0, stream>>>(ybuf, W_out, ytok);
    k9b_proj_out <<<576,   256, 0, stream>>>(ytok, pow_, pob, x, out);
}